// TGAT_4758823764091
// MI455X (gfx1250) — compile-verified
//
#include <hip/hip_runtime.h>
#include <hip/hip_bf16.h>

typedef __attribute__((ext_vector_type(2))) float v2f;
typedef __attribute__((ext_vector_type(8))) float v8f;

#define HEADS 4
#define C1 64      // hidden per head (layer 1)
#define C2 64      // output channels (layer 2)
#define TSTEPS 4
#define K2DIM 256  // HEADS*C1, K of layer-2 GEMM

__device__ __forceinline__ float lrelu(float v) { return v > 0.f ? v : 0.2f * v; }

// float atomic max via ordered-integer transform (correct for arbitrary sign mixes)
__device__ __forceinline__ void atomicMaxF(float* addr, float value) {
    if (value >= 0.f) (void)atomicMax((int*)addr, __float_as_int(value));
    else              (void)atomicMin((unsigned int*)addr, __float_as_uint(value));
}

// ---------------------------------------------------------------------------
// K0: tiny precompute.  u = W1·Wf ; v_t = W1·(bf+bt+time_idx[t]·Wt) ;
//     tw = softmax(temporal_attention); p_*[h] = <u_h, a_*1_h>; q_*[t,h] = <v_t_h, a_*1_h>
// ---------------------------------------------------------------------------
__global__ void k0_precompute(const float* __restrict__ time_idx, const float* __restrict__ Wt,
                              const float* __restrict__ bt, const float* __restrict__ Wf,
                              const float* __restrict__ bf, const float* __restrict__ W1,
                              const float* __restrict__ a_src1, const float* __restrict__ a_dst1,
                              const float* __restrict__ tatt,
                              float* __restrict__ u, float* __restrict__ v, float* __restrict__ tw,
                              float* __restrict__ p_src, float* __restrict__ p_dst,
                              float* __restrict__ q_src, float* __restrict__ q_dst) {
    __shared__ float su[K2DIM];
    __shared__ float sv[TSTEPS][K2DIM];
    int hc = threadIdx.x;  // 0..255

    if (hc == 0) {  // softmax over 4 temporal weights
        float m = tatt[0];
        for (int t = 1; t < TSTEPS; ++t) m = fmaxf(m, tatt[t]);
        float e[TSTEPS], s = 0.f;
        for (int t = 0; t < TSTEPS; ++t) { e[t] = __expf(tatt[t] - m); s += e[t]; }
        for (int t = 0; t < TSTEPS; ++t) tw[t] = e[t] / s;
    }

    float tv[TSTEPS];
    for (int t = 0; t < TSTEPS; ++t) tv[t] = time_idx[t];
    float uu = 0.f, vv[TSTEPS] = {0.f, 0.f, 0.f, 0.f};
    for (int f = 0; f < C1; ++f) {
        float w = W1[hc * C1 + f];
        uu += w * Wf[f];
        float base = bf[f] + bt[f];
        float wt = Wt[f];
        for (int t = 0; t < TSTEPS; ++t) vv[t] += w * (base + tv[t] * wt);
    }
    u[hc] = uu; su[hc] = uu;
    for (int t = 0; t < TSTEPS; ++t) { v[t * K2DIM + hc] = vv[t]; sv[t][hc] = vv[t]; }
    __syncthreads();

    if (hc < HEADS) {
        int h = hc; float ps = 0.f, pd = 0.f;
        for (int c = 0; c < C1; ++c) {
            ps += su[h * C1 + c] * a_src1[h * C1 + c];
            pd += su[h * C1 + c] * a_dst1[h * C1 + c];
        }
        p_src[h] = ps; p_dst[h] = pd;
    }
    if (hc < TSTEPS * HEADS) {
        int t = hc >> 2, h = hc & 3; float qs = 0.f, qd = 0.f;
        for (int c = 0; c < C1; ++c) {
            qs += sv[t][h * C1 + c] * a_src1[h * C1 + c];
            qd += sv[t][h * C1 + c] * a_dst1[h * C1 + c];
        }
        q_src[hc] = qs; q_dst[hc] = qd;
    }
}

// ---------------------------------------------------------------------------
// K1: init layer-1 segment stats. M1 starts at the self-loop logit (every node
//     has a self loop in PyG GATConv), D1/S1 at zero.
// ---------------------------------------------------------------------------
__global__ void k1_init1(const float* __restrict__ x,
                         const float* __restrict__ p_src, const float* __restrict__ p_dst,
                         const float* __restrict__ q_src, const float* __restrict__ q_dst,
                         float* __restrict__ M1, float* __restrict__ D1, float* __restrict__ S1,
                         int N) {
    int tid = blockIdx.x * blockDim.x + threadIdx.x;
    if (tid >= N * 16) return;
    int n = tid >> 4, j = tid & 15;
    int t = j >> 2, h = j & 3;
    float xt = x[n * TSTEPS + t];
    float e = lrelu(xt * (p_src[h] + p_dst[h]) + q_src[j] + q_dst[j]);
    M1[tid] = e; D1[tid] = 0.f; S1[tid] = 0.f;
}

// K2: layer-1 edge pass A — segment max per (dst, head, t)
__global__ void k2_edge_max1(const float* __restrict__ x, const int* __restrict__ ei, int E,
                             const float* __restrict__ p_src, const float* __restrict__ p_dst,
                             const float* __restrict__ q_src, const float* __restrict__ q_dst,
                             float* __restrict__ M1) {
    int e = blockIdx.x * blockDim.x + threadIdx.x;
    if (e >= E) return;
    int s = ei[e], d = ei[E + e];
    float4 xs4 = ((const float4*)x)[s];
    float4 xd4 = ((const float4*)x)[d];
    float xs[4] = {xs4.x, xs4.y, xs4.z, xs4.w};
    float xd[4] = {xd4.x, xd4.y, xd4.z, xd4.w};
#pragma unroll
    for (int j = 0; j < 16; ++j) {
        int t = j >> 2, h = j & 3;
        float val = lrelu(xs[t] * p_src[h] + q_src[j] + xd[t] * p_dst[h] + q_dst[j]);
        atomicMaxF(&M1[d * 16 + j], val);
    }
}

// K3: layer-1 edge pass B — exp-sum (denominator) and exp-weighted sum of x[s,t]
__global__ void k3_edge_sum1(const float* __restrict__ x, const int* __restrict__ ei, int E,
                             const float* __restrict__ p_src, const float* __restrict__ p_dst,
                             const float* __restrict__ q_src, const float* __restrict__ q_dst,
                             const float* __restrict__ M1,
                             float* __restrict__ D1, float* __restrict__ S1) {
    int e = blockIdx.x * blockDim.x + threadIdx.x;
    if (e >= E) return;
    int s = ei[e], d = ei[E + e];
    float4 xs4 = ((const float4*)x)[s];
    float4 xd4 = ((const float4*)x)[d];
    float xs[4] = {xs4.x, xs4.y, xs4.z, xs4.w};
    float xd[4] = {xd4.x, xd4.y, xd4.z, xd4.w};
#pragma unroll
    for (int j = 0; j < 16; ++j) {
        int t = j >> 2, h = j & 3;
        float val = lrelu(xs[t] * p_src[h] + q_src[j] + xd[t] * p_dst[h] + q_dst[j]);
        float ex = __expf(val - M1[d * 16 + j]);
        atomicAdd(&D1[d * 16 + j], ex);
        atomicAdd(&S1[d * 16 + j], ex * xs[t]);
    }
}

// K4a: add analytic self-loop term, normalize: S1 <- (S1 + ex_self*x)/ (D1 + ex_self)
__global__ void k4a_self1(const float* __restrict__ x,
                          const float* __restrict__ p_src, const float* __restrict__ p_dst,
                          const float* __restrict__ q_src, const float* __restrict__ q_dst,
                          const float* __restrict__ M1, const float* __restrict__ D1,
                          float* __restrict__ S1, int N) {
    int tid = blockIdx.x * blockDim.x + threadIdx.x;
    if (tid >= N * 16) return;
    int n = tid >> 4, j = tid & 15;
    int t = j >> 2, h = j & 3;
    float xt = x[n * TSTEPS + t];
    float eself = lrelu(xt * (p_src[h] + p_dst[h]) + q_src[j] + q_dst[j]);
    float ex = __expf(eself - M1[tid]);
    float den = D1[tid] + ex;
    float num = S1[tid] + ex * xt;
    S1[tid] = num / den;
}

// K4b: combined[n,hc] = sum_t tw[t] * relu(u[hc]*S1[n,t,h] + v_t[hc] + b1[hc])
__global__ void k4b_combined(const float* __restrict__ u, const float* __restrict__ v,
                             const float* __restrict__ tw, const float* __restrict__ b1,
                             const float* __restrict__ S1, float* __restrict__ combined, int N) {
    int tid = blockIdx.x * blockDim.x + threadIdx.x;
    if (tid >= N * K2DIM) return;
    int n = tid >> 8, hc = tid & 255, h = hc >> 6;
    float uh = u[hc], bb = b1[hc];
    float acc = 0.f;
#pragma unroll
    for (int t = 0; t < TSTEPS; ++t) {
        float val = uh * S1[n * 16 + t * 4 + h] + v[t * K2DIM + hc] + bb;
        acc += tw[t] * fmaxf(val, 0.f);
    }
    combined[tid] = acc;
}

// ---------------------------------------------------------------------------
// K5: h2 = combined(N x 256) @ W2^T(256 x 64) via V_WMMA_F32_16X16X4_F32.
// One wave per 16x16 D-tile. Block = 128 threads = 4 waves = 4 N-tiles of one
// M-tile. N=20000, K=256, Nout=64 are all multiples of 16: EXEC stays all-1s.
// Per ISA layouts: A lane(l): m=l&15, holds K = {k0+2*(l>>4), +1} -> float2 load.
//                  B lane(l): n=l&15, holds K = {k0+2*(l>>4), +1} of W2 row n.
//                  D lane(l): n=l&15, vgpr j holds row m = j + 8*(l>>4).
// ---------------------------------------------------------------------------
__global__ void k5_gemm_wmma(const float* __restrict__ combined, const float* __restrict__ W2,
                             float* __restrict__ h2) {
    int mTile = blockIdx.x;            // N/16 blocks
    int nTile = threadIdx.x >> 5;      // 4 waves -> 4 output-column tiles
    int lane = threadIdx.x & 31;
    int hi = lane >> 4;                // 0: low half, 1: high half
    int lm = lane & 15;
    const float* Arow = combined + (mTile * 16 + lm) * K2DIM;  // A row m
    const float* Brow = W2 + (nTile * 16 + lm) * K2DIM;        // B[k][n] = W2[n][k]
    v8f acc = {0.f, 0.f, 0.f, 0.f, 0.f, 0.f, 0.f, 0.f};
#pragma unroll 4
    for (int k0 = 0; k0 < K2DIM; k0 += 4) {
        v2f a = *(const v2f*)(Arow + k0 + 2 * hi);
        v2f b = *(const v2f*)(Brow + k0 + 2 * hi);
        acc = __builtin_amdgcn_wmma_f32_16x16x4_f32(false, a, false, b, (short)0, acc,
                                                    false, false);
    }
    float* Dbase = h2 + (mTile * 16 + 8 * hi) * C2 + nTile * 16 + lm;
#pragma unroll
    for (int j = 0; j < 8; ++j) Dbase[j * C2] = acc[j];
}

// K6: layer-2 logits per node; init M2 with self-loop logit, D2 = 0, zero d_out
__global__ void k6_attn2(const float* __restrict__ h2, const float* __restrict__ a_src2,
                         const float* __restrict__ a_dst2, float* __restrict__ als2,
                         float* __restrict__ ald2, float* __restrict__ M2,
                         float* __restrict__ D2, float* __restrict__ out, int N) {
    int n = blockIdx.x * blockDim.x + threadIdx.x;
    if (n >= N) return;
    float as = 0.f, ad = 0.f;
    for (int c = 0; c < C2; ++c) {
        float hv = h2[n * C2 + c];
        as += hv * a_src2[c];
        ad += hv * a_dst2[c];
        out[n * C2 + c] = 0.f;
    }
    als2[n] = as; ald2[n] = ad;
    M2[n] = lrelu(as + ad);
    D2[n] = 0.f;
}

// K7: layer-2 edge pass A — segment max
__global__ void k7_edge_max2(const int* __restrict__ ei, int E, const float* __restrict__ als2,
                             const float* __restrict__ ald2, float* __restrict__ M2) {
    int e = blockIdx.x * blockDim.x + threadIdx.x;
    if (e >= E) return;
    int s = ei[e], d = ei[E + e];
    atomicMaxF(&M2[d], lrelu(als2[s] + ald2[d]));
}

// K8: layer-2 edge scatter — one thread per (edge, channel): coalesced gather of
//     h2[s,:], coalesced atomic adds to out[d,:]; lane c==0 also adds denominator.
__global__ void k8_edge_scatter2(const int* __restrict__ ei, int E,
                                 const float* __restrict__ als2, const float* __restrict__ ald2,
                                 const float* __restrict__ M2, const float* __restrict__ h2,
                                 float* __restrict__ D2, float* __restrict__ out) {
    long long gid = (long long)blockIdx.x * blockDim.x + threadIdx.x;
    int e = (int)(gid >> 6);
    int c = (int)(gid & 63);
    if (e >= E) return;
    int s = ei[e], d = ei[E + e];
    float ex = __expf(lrelu(als2[s] + ald2[d]) - M2[d]);
    if (c == 0) atomicAdd(&D2[d], ex);
    atomicAdd(&out[d * C2 + c], ex * h2[s * C2 + c]);
}

// K9: finalize — add self-loop term locally, divide, add bias
__global__ void k9_final(const float* __restrict__ als2, const float* __restrict__ ald2,
                         const float* __restrict__ M2, const float* __restrict__ D2,
                         const float* __restrict__ h2, const float* __restrict__ b2,
                         float* __restrict__ out, int N) {
    int tid = blockIdx.x * blockDim.x + threadIdx.x;
    if (tid >= N * C2) return;
    int n = tid >> 6, c = tid & 63;
    float ex = __expf(lrelu(als2[n] + ald2[n]) - M2[n]);
    float num = out[tid] + ex * h2[n * C2 + c];
    float den = D2[n] + ex;
    out[tid] = num / den + b2[c];
}

extern "C" void kernel_launch(void* const* d_in, const int* in_sizes, int n_in,
                              void* d_out, int out_size, void* d_ws, size_t ws_size,
                              hipStream_t stream) {
    const float* x       = (const float*)d_in[0];
    const int*   ei      = (const int*)d_in[1];
    const float* time_id = (const float*)d_in[2];
    const float* Wt      = (const float*)d_in[3];
    const float* bt      = (const float*)d_in[4];
    const float* Wf      = (const float*)d_in[5];
    const float* bf      = (const float*)d_in[6];
    const float* W1      = (const float*)d_in[7];
    const float* a_src1  = (const float*)d_in[8];
    const float* a_dst1  = (const float*)d_in[9];
    const float* b1      = (const float*)d_in[10];
    const float* tatt    = (const float*)d_in[11];
    const float* W2      = (const float*)d_in[12];
    const float* a_src2  = (const float*)d_in[13];
    const float* a_dst2  = (const float*)d_in[14];
    const float* b2      = (const float*)d_in[15];
    float* out = (float*)d_out;
    float* ws  = (float*)d_ws;

    const int N = in_sizes[0] / TSTEPS;  // 20000
    const int E = in_sizes[1] / 2;       // 320000

    // workspace layout (floats)
    float* u      = ws;                    // 256
    float* v      = u + 256;               // 4*256
    float* tw     = v + 4 * 256;           // 4
    float* p_src  = tw + 4;                // 4
    float* p_dst  = p_src + 4;             // 4
    float* q_src  = p_dst + 4;             // 16
    float* q_dst  = q_src + 16;            // 16
    float* M1     = ws + 1536;             // N*16
    float* D1     = M1 + (size_t)N * 16;   // N*16
    float* S1     = D1 + (size_t)N * 16;   // N*16
    float* comb   = S1 + (size_t)N * 16;   // N*256
    float* h2     = comb + (size_t)N * K2DIM;  // N*64
    float* als2   = h2 + (size_t)N * C2;   // N
    float* ald2   = als2 + N;              // N
    float* M2     = ald2 + N;              // N
    float* D2     = M2 + N;                // N

    const int B = 256;
    k0_precompute<<<1, 256, 0, stream>>>(time_id, Wt, bt, Wf, bf, W1, a_src1, a_dst1, tatt,
                                         u, v, tw, p_src, p_dst, q_src, q_dst);
    k1_init1<<<(N * 16 + B - 1) / B, B, 0, stream>>>(x, p_src, p_dst, q_src, q_dst,
                                                     M1, D1, S1, N);
    k2_edge_max1<<<(E + B - 1) / B, B, 0, stream>>>(x, ei, E, p_src, p_dst, q_src, q_dst, M1);
    k3_edge_sum1<<<(E + B - 1) / B, B, 0, stream>>>(x, ei, E, p_src, p_dst, q_src, q_dst,
                                                    M1, D1, S1);
    k4a_self1<<<(N * 16 + B - 1) / B, B, 0, stream>>>(x, p_src, p_dst, q_src, q_dst,
                                                      M1, D1, S1, N);
    k4b_combined<<<(N * K2DIM + B - 1) / B, B, 0, stream>>>(u, v, tw, b1, S1, comb, N);
    // WMMA GEMM: one block (4 waves) per 16-row tile; N assumed multiple of 16 (20000 is).
    k5_gemm_wmma<<<N / 16, 128, 0, stream>>>(comb, W2, h2);
    k6_attn2<<<(N + B - 1) / B, B, 0, stream>>>(h2, a_src2, a_dst2, als2, ald2, M2, D2, out, N);
    k7_edge_max2<<<(E + B - 1) / B, B, 0, stream>>>(ei, E, als2, ald2, M2);
    long long scatter_threads = (long long)E * C2;
    k8_edge_scatter2<<<(int)((scatter_threads + B - 1) / B), B, 0, stream>>>(
        ei, E, als2, ald2, M2, h2, D2, out);
    k9_final<<<(N * C2 + B - 1) / B, B, 0, stream>>>(als2, ald2, M2, D2, h2, b2, out, N);
}